// PatchEncoder_14774687498517
// MI455X (gfx1250) — compile-verified
//
#include <hip/hip_runtime.h>
#include <math.h>
#include <stdint.h>

// Problem constants (from reference setup_inputs)
#define B_    128
#define P_    196
#define A_    768   // also D
#define MMAX  147   // floor(196*0.75)
#define UMAX  49    // 196-147
#define MU    (B_ * UMAX)   // 6272 gathered GEMM rows

typedef __attribute__((ext_vector_type(2))) float v2f;
typedef __attribute__((ext_vector_type(8))) float v8f;

// ---------------------------------------------------------------------------
// CDNA5 async global->LDS copy (tracked by ASYNCcnt). 16B per lane.
// Generic __shared__ pointers truncate to the wave-relative LDS offset.
// ---------------------------------------------------------------------------
__device__ __forceinline__ void async_copy_b128(void* lds_ptr, const void* gptr)
{
    asm volatile("global_load_async_to_lds_b128 %0, %1, off"
                 :: "v"((unsigned)(uintptr_t)lds_ptr),
                    "v"((unsigned long long)(uintptr_t)gptr)
                 : "memory");
}
__device__ __forceinline__ void wait_async_zero()
{
    asm volatile("s_wait_asynccnt 0x0" ::: "memory");
}

// ---------------------------------------------------------------------------
// Kernel 1: per-batch-row index computation (top-k masking logic).
// One block per batch row. Reproduces jax top_k (descending, stable) and the
// stable argsort of indicator*P + arange exactly.
// ---------------------------------------------------------------------------
__global__ __launch_bounds__(256) void indices_kernel(
    const float* __restrict__ patches,
    const float* __restrict__ rscores,
    int*   __restrict__ ws_row_src,      // [MU]  = b*196 + unmask_idx
    int*   __restrict__ ws_mask_idx,     // [B_*MMAX]
    int*   __restrict__ ws_unmask_idx,   // [B_*UMAX]
    int*   __restrict__ ws_mask_counts,  // [B_]
    float* __restrict__ out_mask_f,      // d_out section 4 (mask_idx as float)
    float* __restrict__ out_unmask_f)    // d_out section 5 (unmask_idx as float)
{
    __shared__ float validf[P_];
    __shared__ float scores[P_];
    __shared__ int   order[P_];
    __shared__ int   indicator[P_];
    __shared__ int   unm[UMAX];
    __shared__ int   sh_mc;
    __shared__ int   sh_hasv;

    const int b    = blockIdx.x;
    const int tid  = threadIdx.x;
    const int wave = tid >> 5;
    const int lane = tid & 31;

    // patch validity: mean(|x|) > 0  <=>  sum(|x|) > 0  (wave32 reduction)
    for (int p = wave; p < P_; p += 8) {
        const float* row = patches + ((size_t)b * P_ + p) * A_;
        float s = 0.f;
        for (int d = lane; d < A_; d += 32) s += fabsf(row[d]);
        for (int off = 16; off > 0; off >>= 1) s += __shfl_down(s, off, 32);
        if (lane == 0) validf[p] = (s > 0.f) ? 1.f : 0.f;
    }
    __syncthreads();

    if (tid < P_)
        scores[tid] = rscores[(size_t)b * P_ + tid] * 1000.0f * validf[tid];
    __syncthreads();

    // stable descending rank == jax.lax.top_k(scores, P) index order
    if (tid < P_) {
        const float si = scores[tid];
        int r = 0;
        for (int j2 = 0; j2 < P_; ++j2) {
            const float sj = scores[j2];
            r += (sj > si) || (sj == si && j2 < tid);
        }
        order[r] = tid;
        indicator[tid] = 0;
    }
    if (tid == 0) {
        float vc = 0.f;
        for (int p = 0; p < P_; ++p) vc += validf[p];
        const int ivc  = (int)vc;
        const int minu = (int)ceilf(vc * 0.25f);
        int mc = (int)floorf(vc * 0.75f);
        const int cap = ivc - minu;
        if (mc > cap) mc = cap;
        if (mc < 0) mc = 0;
        sh_mc = mc;
        sh_hasv = (vc > 0.f) ? 1 : 0;
        ws_mask_counts[b] = mc;
    }
    __syncthreads();

    const int mc = sh_mc, hasv = sh_hasv;
    if (tid < MMAX) {
        const int take = (tid < mc) && hasv;
        const int mi = take ? order[tid] : 0;
        ws_mask_idx[b * MMAX + tid] = mi;
        out_mask_f[b * MMAX + tid]  = (float)mi;
        if (take) indicator[mi] = 1;   // distinct mi values -> race free
    }
    __syncthreads();

    // stable argsort(indicator*P + arange)[:UMAX] == first UMAX p with
    // indicator==0, ascending (also matches the !has_valid special case).
    if (tid == 0) {
        int cnt = 0;
        for (int p = 0; p < P_ && cnt < UMAX; ++p)
            if (!indicator[p]) unm[cnt++] = p;
    }
    __syncthreads();

    if (tid < UMAX) {
        const int ui = unm[tid];
        ws_unmask_idx[b * UMAX + tid] = ui;
        ws_row_src[b * UMAX + tid]    = b * P_ + ui;
        out_unmask_f[b * UMAX + tid]  = (float)ui;
    }
}

// ---------------------------------------------------------------------------
// Kernel 2: gathered GEMM for unmasked embeddings using f32 WMMA.
// C[m, n] = patches[row_src[m], :] @ W[:, n] + bias[n] + pos[unmask_idx[m], n]
// Block: 256 threads = 8 waves (2 in M x 4 in N). Block tile 128(M) x 128(N).
// Wave tile 64x32 = 4x2 V_WMMA_F32_16X16X4_F32 accumulators.
// K staged in 16-chunks via double-buffered async global->LDS copies.
// ---------------------------------------------------------------------------
#define TM 128
#define TN 128
#define TK 16
#define APITCH 20    // 80B rows: 16B aligned; 20L mod 64 distinct for L=0..15
#define BPITCH 132   // 528B rows: 16B aligned

__global__ __launch_bounds__(256) void gemm_unmasked_kernel(
    const float* __restrict__ patches,
    const float* __restrict__ W,
    const float* __restrict__ bias,
    const float* __restrict__ pos,
    const int*   __restrict__ row_src,     // [MU] = b*196 + idx
    const int*   __restrict__ unmask_idx,  // [MU] = idx (pos row)
    float*       __restrict__ out_u)       // [MU][A_]
{
    __shared__ __align__(16) float As[2][TM][APITCH];
    __shared__ __align__(16) float Bs[2][TK][BPITCH];

    const int tid     = threadIdx.x;
    const int wave    = tid >> 5;
    const int lane    = tid & 31;
    const int m_block = blockIdx.x * TM;   // 49 blocks
    const int n_block = blockIdx.y * TN;   // 6 blocks
    const int wm = (wave >> 2) << 6;       // 0,64
    const int wn = (wave & 3) << 5;        // 0,32,64,96
    const int ml    = lane & 15;
    const int khalf = (lane >> 4) << 1;    // lanes 0-15: K+0/1, lanes 16-31: K+2/3

    // A async slots: 128 rows x 4 float4 = 512 / 256 threads = 2 per thread
    const float* aG[2];
    float* aL[2][2];
    #pragma unroll
    for (int i = 0; i < 2; ++i) {
        const int f  = tid + (i << 8);
        const int r  = f >> 2;
        const int c4 = f & 3;
        const int src = row_src[m_block + r];
        aG[i] = patches + (size_t)src * A_ + (c4 << 2);
        aL[0][i] = &As[0][r][c4 << 2];
        aL[1][i] = &As[1][r][c4 << 2];
    }
    // B async slots: 16 k-rows x 32 float4 = 512 / 256 threads = 2 per thread
    const float* bG[2];
    float* bL[2][2];
    #pragma unroll
    for (int i = 0; i < 2; ++i) {
        const int f  = tid + (i << 8);
        const int kr = f >> 5;
        const int c4 = f & 31;
        bG[i] = W + (size_t)kr * A_ + n_block + (c4 << 2);
        bL[0][i] = &Bs[0][kr][c4 << 2];
        bL[1][i] = &Bs[1][kr][c4 << 2];
    }

    v8f acc[4][2] = {};

    // prologue: chunk 0 into buffer 0
    #pragma unroll
    for (int i = 0; i < 2; ++i) async_copy_b128(aL[0][i], aG[i]);
    #pragma unroll
    for (int i = 0; i < 2; ++i) async_copy_b128(bL[0][i], bG[i]);

    for (int k0 = 0; k0 < A_; k0 += TK) {
        const int cur = (k0 >> 4) & 1;
        wait_async_zero();      // this chunk's LDS data complete (own wave)
        __syncthreads();        // ...and everyone's; buf^1 free of readers
        if (k0 + TK < A_) {
            const int nxt = cur ^ 1;
            const int kn  = k0 + TK;
            #pragma unroll
            for (int i = 0; i < 2; ++i) async_copy_b128(aL[nxt][i], aG[i] + kn);
            #pragma unroll
            for (int i = 0; i < 2; ++i) async_copy_b128(bL[nxt][i], bG[i] + (size_t)kn * A_);
        }

        #pragma unroll
        for (int kk = 0; kk < TK; kk += 4) {
            // A frag 16x4: lane L<16 -> M=L,K=kk+0/1 ; L>=16 -> M=L-16,K=kk+2/3
            v2f a[4];
            #pragma unroll
            for (int i = 0; i < 4; ++i)
                a[i] = *(const v2f*)&As[cur][wm + (i << 4) + ml][kk + khalf];
            // B frag 4x16: lane holds N=ml, same K pairing across lane halves
            v2f bf[2];
            #pragma unroll
            for (int j = 0; j < 2; ++j) {
                bf[j].x = Bs[cur][kk + khalf][wn + (j << 4) + ml];
                bf[j].y = Bs[cur][kk + khalf + 1][wn + (j << 4) + ml];
            }
            #pragma unroll
            for (int i = 0; i < 4; ++i) {
                acc[i][0] = __builtin_amdgcn_wmma_f32_16x16x4_f32(false, a[i], false, bf[0], (short)0, acc[i][0], false, false);
                acc[i][1] = __builtin_amdgcn_wmma_f32_16x16x4_f32(false, a[i], false, bf[1], (short)0, acc[i][1], false, false);
            }
        }
    }

    // Epilogue: D layout — VGPR v: lanes0-15 -> M=v, lanes16-31 -> M=v+8; N=lane%16
    const int rbase = (lane >> 4) << 3;
    #pragma unroll
    for (int j = 0; j < 2; ++j) {
        const int n = n_block + wn + (j << 4) + ml;
        const float bj = bias[n];
        #pragma unroll
        for (int i = 0; i < 4; ++i) {
            #pragma unroll
            for (int v = 0; v < 8; ++v) {
                const int m = m_block + wm + (i << 4) + rbase + v;
                const int p = unmask_idx[m];   // original patch index for pos
                out_u[(size_t)m * A_ + n] = acc[i][j][v] + bj + pos[(size_t)p * A_ + n];
            }
        }
    }
}

// ---------------------------------------------------------------------------
// Kernel 3: mt = mask_token @ W + b   (1x768 GEMV — trivial)
// ---------------------------------------------------------------------------
__global__ void mask_token_kernel(const float* __restrict__ mask_token,
                                  const float* __restrict__ W,
                                  const float* __restrict__ bias,
                                  float* __restrict__ mt)
{
    const int d = blockIdx.x * blockDim.x + threadIdx.x;
    if (d >= A_) return;
    float s = bias[d];
    for (int a = 0; a < A_; ++a)
        s = fmaf(mask_token[a], W[(size_t)a * A_ + d], s);
    mt[d] = s;
}

// ---------------------------------------------------------------------------
// Kernel 4a: masked_embeddings = (mt + pos[mask_idx]) * (j < mask_count)
// ---------------------------------------------------------------------------
__global__ void masked_fill_kernel(const float* __restrict__ mt,
                                   const float* __restrict__ pos,
                                   const int*   __restrict__ mask_idx,
                                   const int*   __restrict__ mask_counts,
                                   float*       __restrict__ out_m)
{
    const size_t e = (size_t)blockIdx.x * blockDim.x + threadIdx.x;
    const size_t total = (size_t)B_ * MMAX * A_;
    if (e >= total) return;
    const int d = (int)(e % A_);
    const int t = (int)(e / A_);
    const int b = t / MMAX;
    const int j = t % MMAX;
    float v = 0.f;
    if (j < mask_counts[b])
        v = mt[d] + pos[(size_t)mask_idx[t] * A_ + d];
    out_m[e] = v;
}

// ---------------------------------------------------------------------------
// Kernel 4b: unmasked_positions = pos[unmask_idx]
// ---------------------------------------------------------------------------
__global__ void unmasked_pos_kernel(const float* __restrict__ pos,
                                    const int*   __restrict__ unmask_idx,
                                    float*       __restrict__ out_up)
{
    const size_t e = (size_t)blockIdx.x * blockDim.x + threadIdx.x;
    const size_t total = (size_t)B_ * UMAX * A_;
    if (e >= total) return;
    const int d = (int)(e % A_);
    const int t = (int)(e / A_);
    out_up[e] = pos[(size_t)unmask_idx[t] * A_ + d];
}

// ---------------------------------------------------------------------------
extern "C" void kernel_launch(void* const* d_in, const int* in_sizes, int n_in,
                              void* d_out, int out_size, void* d_ws, size_t ws_size,
                              hipStream_t stream)
{
    const float* patches = (const float*)d_in[0];
    const float* rscores = (const float*)d_in[1];
    const float* W       = (const float*)d_in[2];
    const float* bias    = (const float*)d_in[3];
    const float* pos     = (const float*)d_in[4];
    const float* mtok    = (const float*)d_in[5];
    float* out = (float*)d_out;

    // Workspace carve-up (~127 KB)
    char* ws = (char*)d_ws;
    int* ws_row_src     = (int*)ws;  ws += sizeof(int) * MU;
    int* ws_mask_idx    = (int*)ws;  ws += sizeof(int) * B_ * MMAX;
    int* ws_unmask_idx  = (int*)ws;  ws += sizeof(int) * MU;
    int* ws_mask_counts = (int*)ws;  ws += sizeof(int) * B_;
    float* ws_mt        = (float*)ws;

    // Output layout (flat floats, reference return order)
    const size_t OUT_U  = 0;
    const size_t OUT_M  = (size_t)B_ * UMAX * A_;                 // after unmasked_embeddings
    const size_t OUT_UP = OUT_M  + (size_t)B_ * MMAX * A_;        // after masked_embeddings
    const size_t OUT_MI = OUT_UP + (size_t)B_ * UMAX * A_;        // after unmasked_positions
    const size_t OUT_UI = OUT_MI + (size_t)B_ * MMAX;             // after mask_idx

    indices_kernel<<<B_, 256, 0, stream>>>(patches, rscores, ws_row_src,
        ws_mask_idx, ws_unmask_idx, ws_mask_counts, out + OUT_MI, out + OUT_UI);

    mask_token_kernel<<<(A_ + 255) / 256, 256, 0, stream>>>(mtok, W, bias, ws_mt);

    dim3 g2(MU / TM, A_ / TN);   // 49 x 6
    gemm_unmasked_kernel<<<g2, 256, 0, stream>>>(patches, W, bias, pos,
                                                 ws_row_src, ws_unmask_idx,
                                                 out + OUT_U);

    {
        const size_t total = (size_t)B_ * MMAX * A_;
        masked_fill_kernel<<<(unsigned)((total + 255) / 256), 256, 0, stream>>>(
            ws_mt, pos, ws_mask_idx, ws_mask_counts, out + OUT_M);
    }
    {
        const size_t total = (size_t)B_ * UMAX * A_;
        unmasked_pos_kernel<<<(unsigned)((total + 255) / 256), 256, 0, stream>>>(
            pos, ws_unmask_idx, out + OUT_UP);
    }
}